// BilinearInterpolation_85461259255915
// MI455X (gfx1250) — compile-verified
//
#include <hip/hip_runtime.h>

typedef __attribute__((ext_vector_type(2))) float v2f;
typedef __attribute__((ext_vector_type(4))) float v4f;
typedef __attribute__((ext_vector_type(8))) float v8f;

#define BDIM 32
#define HDIM 512
#define WDIM 512
#define CDIM 8   // 8 floats per texel = 2 x float4

// One wave32 handles 16 consecutive pixels of one image row.
// Lane l: pixel p = l>>1, channel-group cg = l&1 (4 floats).
// The affine grid transform (the reference's einsum) runs on the matrix core:
//   D(16x16) = A(16x4: padded theta) x B(4x16: grid columns [xs, ys, 1, 0])
// giving x_n in D[0] lane n, y_n in D[1] lane n.
__global__ __launch_bounds__(256) void stn_bilinear_wmma(
    const float* __restrict__ X, const float* __restrict__ theta,
    float* __restrict__ out)
{
    const int tid = blockIdx.x * blockDim.x + threadIdx.x;
    const int ll  = threadIdx.x & 31;     // lane within wave32
    const int wv  = tid >> 5;             // global wave id
    const int pb  = wv << 4;              // first of this wave's 16 pixels

    // Pixel decomposition; wave never crosses a row (512 % 16 == 0).
    const int j0  = pb & (WDIM - 1);
    const int row = pb >> 9;              // pb / W
    const int i   = row & (HDIM - 1);
    const int b   = row >> 9;             // row / H

    // ---- theta: wave-uniform batch -> scalar loads ----
    const int bu = __builtin_amdgcn_readfirstlane(b);
    const float* __restrict__ t = theta + bu * 6;
    const float t0 = t[0], t1 = t[1], t2 = t[2];
    const float t3 = t[3], t4 = t[4], t5 = t[5];

    // ---- A matrix (16x4 f32): row0 = [t0 t1 t2 0], row1 = [t3 t4 t5 0], rest 0.
    // Layout: VGPR0 = K0 (lanes 0-15) / K2 (lanes 16-31); VGPR1 = K1 / K3.
    v2f a;
    a.x = (ll == 0) ? t0 : (ll == 1) ? t3 : (ll == 16) ? t2 : (ll == 17) ? t5 : 0.0f;
    a.y = (ll == 0) ? t1 : (ll == 1) ? t4 : 0.0f;

    // ---- B matrix (4x16 f32): column n = [xs_n, ys, 1, 0].
    const float ys = -1.0f + 2.0f * (float)i * (1.0f / (float)(HDIM - 1));
    const int   n  = ll & 15;
    const float xs = -1.0f + 2.0f * (float)(j0 + n) * (1.0f / (float)(WDIM - 1));
    v2f bm;
    bm.x = (ll < 16) ? xs : 1.0f;   // K0 rows / K2 rows
    bm.y = (ll < 16) ? ys : 0.0f;   // K1 rows / K3 rows

    v8f c = {};
    v8f d = __builtin_amdgcn_wmma_f32_16x16x4_f32(
        /*neg_a=*/false, a, /*neg_b=*/false, bm,
        /*c_mod=*/(short)0, c, /*reuse_a=*/false, /*reuse_b=*/false);

    // ---- broadcast coords: lane l needs pixel p = l>>1 from lane p ----
    const int p  = ll >> 1;
    const int cg = ll & 1;
    union { float f; int i_; } ux, uy, rx, ry;
    ux.f = d[0];  uy.f = d[1];
    rx.i_ = __builtin_amdgcn_ds_bpermute(p << 2, ux.i_);
    ry.i_ = __builtin_amdgcn_ds_bpermute(p << 2, uy.i_);

    // ---- map to image space, exactly as the reference ----
    const float xf = 0.5f * (rx.f + 1.0f) * (float)WDIM;
    const float yf = 0.5f * (ry.f + 1.0f) * (float)HDIM;

    int x0 = (int)floorf(xf), y0 = (int)floorf(yf);
    int x1 = x0 + 1,          y1 = y0 + 1;
    x0 = min(max(x0, 0), WDIM - 1);
    x1 = min(max(x1, 0), WDIM - 1);
    y0 = min(max(y0, 0), HDIM - 1);
    y1 = min(max(y1, 0), HDIM - 1);
    const float x0f = (float)x0, x1f = (float)x1;
    const float y0f = (float)y0, y1f = (float)y1;
    const float wa = (x1f - xf) * (y1f - yf);
    const float wb = (x1f - xf) * (yf - y0f);
    const float wc = (xf - x0f) * (y1f - yf);
    const float wd = (xf - x0f) * (yf - y0f);

    // ---- 4-corner gather, 128-bit each (texel = 32B, 16B-aligned halves) ----
    const v4f* __restrict__ X4 = (const v4f*)X;
    const int rb  = b * (HDIM * WDIM);
    const int ia  = ((rb + y0 * WDIM + x0) << 1) + cg;
    const int ib  = ((rb + y1 * WDIM + x0) << 1) + cg;
    const int ic  = ((rb + y0 * WDIM + x1) << 1) + cg;
    const int idd = ((rb + y1 * WDIM + x1) << 1) + cg;
    const v4f Ia = X4[ia];
    const v4f Ib = X4[ib];
    const v4f Ic = X4[ic];
    const v4f Id = X4[idd];

    v4f o;
    o.x = wa * Ia.x + wb * Ib.x + wc * Ic.x + wd * Id.x;
    o.y = wa * Ia.y + wb * Ib.y + wc * Ic.y + wd * Id.y;
    o.z = wa * Ia.z + wb * Ib.z + wc * Ic.z + wd * Id.z;
    o.w = wa * Ia.w + wb * Ib.w + wc * Ic.w + wd * Id.w;

    // ---- streaming store: lane l writes floats [tid*4 .. tid*4+3] -> fully
    // coalesced 512B per wave; nontemporal so output doesn't evict input in L2.
    v4f* O4 = (v4f*)out;
    __builtin_nontemporal_store(o, O4 + tid);
}

extern "C" void kernel_launch(void* const* d_in, const int* in_sizes, int n_in,
                              void* d_out, int out_size, void* d_ws, size_t ws_size,
                              hipStream_t stream) {
    const float* X     = (const float*)d_in[0];   // (32, 512, 512, 8) f32
    const float* theta = (const float*)d_in[1];   // (32, 6) f32
    float* out = (float*)d_out;                   // (32, 512, 512, 8) f32

    // B*H*W pixels, 2 channel-groups each -> one thread per (pixel, cg)
    const long long total = (long long)BDIM * HDIM * WDIM * 2;  // 16,777,216
    const int block = 256;
    const int grid  = (int)(total / block);                      // 65,536
    stn_bilinear_wmma<<<grid, block, 0, stream>>>(X, theta, out);
}